// SparseDownSample_76845554860272
// MI455X (gfx1250) — compile-verified
//
#include <hip/hip_runtime.h>
#include <hip/hip_bf16.h>

// Problem constants (fixed by the reference setup).
constexpr int B  = 8;
constexpr int C  = 16;
constexpr int H  = 352;
constexpr int W  = 1216;
constexpr int H2 = H / 2;   // 176
constexpr int W2 = W / 2;   // 608

typedef float v2f __attribute__((ext_vector_type(2)));
typedef __attribute__((address_space(3))) float lds_float;

// Block = 32x8 output pixels (one wave per output row-slice), grid = (19,22,8).
// Per block: async-stage the 16x64 depth input tile (4KB) into LDS while all
// geo candidate loads stream in NT; then argmax + in-register select.
__global__ __launch_bounds__(256) void sparse_downsample_kernel(
    const float* __restrict__ depth,
    const float* __restrict__ geo,
    float* __restrict__ out)
{
    __shared__ float sdepth[16 * 64];   // 16 input rows x 64 input cols

    const int tx = threadIdx.x;         // 0..31  -> output col within tile
    const int ty = threadIdx.y;         // 0..7   -> output row within tile
    const int bx = blockIdx.x;          // 0..18
    const int by = blockIdx.y;          // 0..21
    const int b  = blockIdx.z;          // batch

    const int l = ty * 32 + tx;         // 0..255 linear tid

    // ---- 1) Fire-and-forget: stage depth tile into LDS via the CDNA5 async
    //         copy path (ASYNCcnt). Each thread moves 16 bytes (b128).
    {
        const int tr = l >> 4;                 // tile row 0..15
        const int tc = (l & 15) << 2;          // float col 0,4,...,60
        const float* gsrc = depth
            + ((size_t)b * H + (size_t)by * 16 + tr) * W
            + (size_t)bx * 64 + tc;
        // Raw LDS byte offset: ptrtoint of an addrspace(3) pointer.
        unsigned lds_dst =
            (unsigned)(size_t)((lds_float*)sdepth + (l << 2));
        asm volatile("global_load_async_to_lds_b128 %0, %1, off"
                     :: "v"(lds_dst), "v"(gsrc)
                     : "memory");
    }

    const int oy = by * 8 + ty;
    const int ox = bx * 32 + tx;

    // ---- 2) Issue ALL geo candidate loads now (addresses independent of the
    //         argmax), non-temporal: 219MB streams through once, keep it out
    //         of L2's way. Two b64 loads per channel = both 2x2 rows.
    v2f g0[C];  // row 2*oy,   cols 2*ox, 2*ox+1
    v2f g1[C];  // row 2*oy+1, cols 2*ox, 2*ox+1
    const size_t p0 = ((size_t)(b * C) * H + (size_t)(2 * oy)) * W + (size_t)(2 * ox);
    #pragma unroll
    for (int c = 0; c < C; ++c) {
        const size_t pc = p0 + (size_t)c * ((size_t)H * W);
        g0[c] = __builtin_nontemporal_load((const v2f*)(geo + pc));
        g1[c] = __builtin_nontemporal_load((const v2f*)(geo + pc + W));
    }

    // ---- 3) Drain our async copy, make the tile visible block-wide.
    asm volatile("s_wait_asynccnt 0" ::: "memory");
    __syncthreads();

    // Depth 2x2 window from LDS (8B-aligned ds_load_b64, conflict-free:
    // lane tx hits banks 2*tx, 2*tx+1 across 64 banks).
    const v2f da = *(const v2f*)&sdepth[(2 * ty)     * 64 + 2 * tx];
    const v2f db = *(const v2f*)&sdepth[(2 * ty + 1) * 64 + 2 * tx];

    // First-occurrence argmax over [da.x, da.y, db.x, db.y] (strict >).
    float best = da.x; int idx = 0;
    if (da.y > best) { best = da.y; idx = 1; }
    if (db.x > best) { best = db.x; idx = 2; }
    if (db.y > best) { best = db.y; idx = 3; }

    // ---- 4) Stores (NT: write-once streams). Output layout: depth_fold
    //         [B,1,H2,W2] followed by geo_fold [B,C,H2,W2].
    __builtin_nontemporal_store(best, out + ((size_t)b * H2 + oy) * W2 + ox);

    const bool rsel = (idx >= 2);
    const bool csel = (idx & 1);
    const size_t geo_out_base = (size_t)B * H2 * W2;
    #pragma unroll
    for (int c = 0; c < C; ++c) {
        const v2f row = rsel ? g1[c] : g0[c];
        const float sel = csel ? row.y : row.x;
        __builtin_nontemporal_store(
            sel,
            out + geo_out_base + (((size_t)(b * C + c)) * H2 + oy) * W2 + ox);
    }
}

extern "C" void kernel_launch(void* const* d_in, const int* in_sizes, int n_in,
                              void* d_out, int out_size, void* d_ws, size_t ws_size,
                              hipStream_t stream) {
    const float* depth = (const float*)d_in[0];   // [8,1,352,1216] f32
    const float* geo   = (const float*)d_in[1];   // [8,16,352,1216] f32
    float* out         = (float*)d_out;           // depth_fold ++ geo_fold, f32

    dim3 block(32, 8, 1);                 // 256 threads = 8 waves (wave32)
    dim3 grid(W2 / 32, H2 / 8, B);        // (19, 22, 8) — exact tiling
    sparse_downsample_kernel<<<grid, block, 0, stream>>>(depth, geo, out);
}